// NewtonLoss_8916352106646
// MI455X (gfx1250) — compile-verified
//
#include <hip/hip_runtime.h>
#include <hip/hip_bf16.h>
#include <stdint.h>

// NewtonLoss on MI455X (gfx1250): memory-bound streaming (220 MB @ 23.3 TB/s ~ 9.5us).
// Phase 1: TDM-load 8x768f tiles of states_x to LDS, per-atom sq-dist, wave32
//          segmented reduction over sorted molecule_id, atomics at run heads.
// Phase 2: per-molecule means summed across molecules with V_WMMA_F32_16X16X4_F32
//          (A = ones -> D columns accumulate channel sums; cols 0..7 = states,
//          col 8 = valid-molecule count), full f32 precision.
// Phase 3: tiny finalize kernel applies gamma weights.

#define GAMMA_F 0.7f
#define KST 8
#define CHUNK 256   // atoms per block

typedef float v2f  __attribute__((ext_vector_type(2)));
typedef float v8f  __attribute__((ext_vector_type(8)));
typedef unsigned int u32x4 __attribute__((ext_vector_type(4)));
typedef int   i32x4 __attribute__((ext_vector_type(4)));
typedef int   i32x8 __attribute__((ext_vector_type(8)));

#if defined(__has_builtin)
#if __has_builtin(__builtin_amdgcn_tensor_load_to_lds) && __has_builtin(__builtin_amdgcn_s_wait_tensorcnt)
#define USE_TDM 1
#endif
#endif
#ifndef USE_TDM
#define USE_TDM 0
#endif

// ---------------- zero workspace: accum[16] | counts[M] | sums[M*8] ----------
__global__ void nl_zero(float* __restrict__ ws, const int* __restrict__ pM) {
  int M = *pM;
  long long total = 16 + (long long)M * (KST + 1);
  long long i = (long long)blockIdx.x * blockDim.x + threadIdx.x;
  long long stride = (long long)gridDim.x * blockDim.x;
  for (; i < total; i += stride) ws[i] = 0.0f;
}

// ---------------- phase 1: per-atom sqdist + segmented wave reduction --------
__global__ __launch_bounds__(CHUNK) void nl_main(const float* __restrict__ states,
                                                 const float* __restrict__ xt,
                                                 const int*   __restrict__ mol,
                                                 float*       __restrict__ ws,
                                                 const int*   __restrict__ pM,
                                                 int N) {
  __shared__ float lds_s[KST * CHUNK * 3];  // 24 KB: 8 state rows x 768 floats
  int M = *pM;
  float* counts = ws + 16;
  float* sums   = ws + 16 + M;
  int chunkStart = blockIdx.x * CHUNK;

#if USE_TDM
  // One TDM DMA per block: 2D tile, tile_dim0 = 768 f32 (contiguous xyz of 256
  // atoms), tile_dim1 = 8 states, row stride = N*3 elements. OOB rows/cols of
  // the tail chunk read as zero via tensor_dim0 clamp.
  if (threadIdx.x < 32) {  // wave-uniform: only wave 0 issues (TDM ignores EXEC)
    unsigned long long gaddr =
        (unsigned long long)(uintptr_t)(states + (size_t)chunkStart * 3);
    unsigned int laddr = (unsigned int)(uintptr_t)(&lds_s[0]);  // LDS byte offset
    unsigned int td0   = (unsigned int)(N - chunkStart) * 3u;   // remaining elems
    unsigned int tile0 = CHUNK * 3u;                            // 768
    unsigned long long s0 = (unsigned long long)N * 3ull;       // dim0 stride

    u32x4 g0;
    g0[0] = 1u;                                    // count=1, no gather
    g0[1] = laddr;                                 // lds_addr
    g0[2] = (unsigned int)(gaddr & 0xFFFFFFFFull); // global_addr[31:0]
    g0[3] = (unsigned int)((gaddr >> 32) & 0x01FFFFFFull) | (2u << 30); // type=2

    i32x8 g1;
    g1[0] = (int)(2u << 16);                                  // data_size = 4B
    g1[1] = (int)((td0 & 0xFFFFu) << 16);                     // tensor_dim0[15:0]
    g1[2] = (int)(((td0 >> 16) & 0xFFFFu) | ((unsigned)KST << 16)); // dim0 hi | tensor_dim1=8
    g1[3] = (int)(tile0 << 16);                               // tile_dim0=768
    g1[4] = (int)KST;                                         // tile_dim1=8, tile_dim2=0
    g1[5] = (int)(unsigned int)(s0 & 0xFFFFFFFFull);          // tensor_dim0_stride lo
    g1[6] = (int)(unsigned int)((s0 >> 32) & 0xFFFFull);      // stride hi | dim1_stride lo
    g1[7] = 0;

    i32x4 g2; g2[0] = g2[1] = g2[2] = g2[3] = 0;   // group 2 unused (2D tile)
    i32x4 g3; g3[0] = g3[1] = g3[2] = g3[3] = 0;   // group 3 unused (2D tile)
    i32x8 g4; g4[0] = g4[1] = g4[2] = g4[3] = g4[4] = g4[5] = g4[6] = g4[7] = 0;

    // amdgpu-toolchain (clang-23) arity: 6 args, trailing i32 = cache policy
    __builtin_amdgcn_tensor_load_to_lds(g0, g1, g2, g3, g4, 0);
    __builtin_amdgcn_s_wait_tensorcnt(0);
  }
#else
  for (int i = threadIdx.x; i < KST * CHUNK * 3; i += blockDim.x) {
    int k = i / (CHUNK * 3);
    int r = i - k * (CHUNK * 3);
    long long g = (long long)chunkStart * 3 + r;
    lds_s[i] = (g < (long long)N * 3) ? states[(size_t)k * N * 3 + g] : 0.0f;
  }
#endif
  __syncthreads();

  int t = threadIdx.x;
  int a = chunkStart + t;
  bool active = (a < N);
  int m = active ? mol[a] : -1;   // harness: integer inputs as int32
  float tx = 0.f, ty = 0.f, tz = 0.f;
  if (active) {
    tx = xt[(size_t)a * 3 + 0];
    ty = xt[(size_t)a * 3 + 1];
    tz = xt[(size_t)a * 3 + 2];
  }

  float v[KST + 1];
#pragma unroll
  for (int k = 0; k < KST; ++k) {
    float sx = lds_s[k * CHUNK * 3 + t * 3 + 0];
    float sy = lds_s[k * CHUNK * 3 + t * 3 + 1];
    float sz = lds_s[k * CHUNK * 3 + t * 3 + 2];
    float dx = sx - tx, dy = sy - ty, dz = sz - tz;
    v[k] = active ? (dx * dx + dy * dy + dz * dz) : 0.0f;
  }
  v[KST] = active ? 1.0f : 0.0f;  // count channel

  // wave32 segmented suffix-sum over sorted runs of molecule_id
  int lane = t & 31;
#pragma unroll
  for (int off = 1; off < 32; off <<= 1) {
    int mo = __shfl_down(m, off, 32);
    bool ok = (lane + off < 32) && (mo == m);
#pragma unroll
    for (int k = 0; k <= KST; ++k) {
      float vo = __shfl_down(v[k], off, 32);
      if (ok) v[k] += vo;
    }
  }
  int mprev = __shfl_up(m, 1, 32);
  bool head = (lane == 0) || (mprev != m);
  if (head && m >= 0 && m < M) {
#pragma unroll
    for (int k = 0; k < KST; ++k)
      atomicAdd(&sums[(size_t)m * KST + k], v[k]);
    atomicAdd(&counts[m], v[KST]);
  }
}

// ---------------- phase 2: sum means over molecules via WMMA f32 ------------
__device__ __forceinline__ float nl_chan(const float* __restrict__ sums,
                                         const float* __restrict__ counts,
                                         int M, int m, int j) {
  if (m >= M || j > KST) return 0.0f;
  float c = counts[m];
  if (!(c > 0.0f)) return 0.0f;          // invalid molecule contributes nothing
  if (j == KST) return 1.0f;             // valid-count channel
  return sums[(size_t)m * KST + j] / c;  // per-molecule mean for state j
}

__global__ __launch_bounds__(256) void nl_phase2(float* __restrict__ ws,
                                                 const int* __restrict__ pM) {
  int M = *pM;
  float*       accum  = ws;
  const float* counts = ws + 16;
  const float* sums   = ws + 16 + M;

  int lane = threadIdx.x & 31;
  int col  = lane & 15;   // B column = lane % 16 -> channel index
  int half = lane >> 4;
  int waveId = (int)((blockIdx.x * blockDim.x + threadIdx.x) >> 5);
  int nwaves = (int)((gridDim.x * blockDim.x) >> 5);

  v2f aOnes; aOnes.x = 1.0f; aOnes.y = 1.0f;   // A = ones(16x4)
  v8f d = {0.f, 0.f, 0.f, 0.f, 0.f, 0.f, 0.f, 0.f};

  int ngroups = (M + 3) >> 2;  // 4 molecules (K-slots) per WMMA
  for (int g = waveId; g < ngroups; g += nwaves) {
    int mA = g * 4 + half;       // (VGPR0, half) K-slot
    int mB = g * 4 + 2 + half;   // (VGPR1, half) K-slot
    v2f b;
    b.x = nl_chan(sums, counts, M, mA, col);
    b.y = nl_chan(sums, counts, M, mB, col);
    // D[i][j] += sum_k B[k][j]  (A all-ones) -> column j accumulates channel j
    d = __builtin_amdgcn_wmma_f32_16x16x4_f32(false, aOnes, false, b,
                                              (short)0, d, false, false);
  }
  // D row 0 lives in VGPR0, lanes 0..15 (col = lane); all rows identical.
  if (lane <= KST) atomicAdd(&accum[lane], d[0]);
}

// ---------------- phase 3: gamma weighting ----------------------------------
__global__ void nl_final(const float* __restrict__ ws, float* __restrict__ out) {
  const float* accum = ws;
  float nv = accum[KST];
  if (nv < 1.0f) nv = 1.0f;
  float w[KST];
  float g = 1.0f, wsum = 0.0f;
  for (int k = KST - 1; k >= 0; --k) { w[k] = g; g *= GAMMA_F; }
  for (int k = 0; k < KST; ++k) wsum += w[k];
  float r = 0.0f;
  for (int k = 0; k < KST; ++k) r += (w[k] / wsum) * (accum[k] / nv);
  *out = r;
}

extern "C" void kernel_launch(void* const* d_in, const int* in_sizes, int n_in,
                              void* d_out, int out_size, void* d_ws, size_t ws_size,
                              hipStream_t stream) {
  (void)n_in; (void)out_size; (void)ws_size;
  const float* states = (const float*)d_in[0];   // (K, N, 3) f32
  const float* xt     = (const float*)d_in[1];   // (N, 3) f32
  const int*   mol    = (const int*)d_in[2];     // (N,) sorted segment ids
  const int*   pM     = (const int*)d_in[3];     // scalar num_molecules (device)
  float* out = (float*)d_out;
  float* ws  = (float*)d_ws;

  int N = in_sizes[1] / 3;

  nl_zero<<<1024, 256, 0, stream>>>(ws, pM);
  int nblk = (N + CHUNK - 1) / CHUNK;
  nl_main<<<nblk, CHUNK, 0, stream>>>(states, xt, mol, ws, pM, N);
  nl_phase2<<<128, 256, 0, stream>>>(ws, pM);
  nl_final<<<1, 1, 0, stream>>>(ws, out);
}